// MaskSupervisionLoss_64458869178902
// MI455X (gfx1250) — compile-verified
//
#include <hip/hip_runtime.h>

typedef __attribute__((ext_vector_type(2))) float v2f;
typedef __attribute__((ext_vector_type(8))) float v8f;

#define EPS_ 1e-6f
#define B_ 32
#define S_ 8
#define HW_ 65536
#define KC_ 512
#define PITCH_ 520           // 8-bank skew per slot row, keeps float2 8B-aligned
#define CHUNKS_ (HW_ / KC_)  // 128

// ---------------------------------------------------------------- init ------
__global__ void msl_init(float* acc) {
  int i = blockIdx.x * blockDim.x + threadIdx.x;
  if (i < B_ * 256) acc[i] = 0.0f;
}

// ---------------------------------------------------------------- accum -----
// Per (batch, chunk): accumulate the augmented 16x16 Gram matrix
//   A = [pred slots 0..7 ; ones ; 0...]   (16 x K)
//   B = [gt   slots 0..7 ; ones ; 0...]^T (K x 16)
// so C[p][g]=inter, C[p][8]=pred sums, C[8][g]=gt sums, C[0][0]=bg inter.
__global__ __launch_bounds__(256) void msl_accum(const float* __restrict__ pred,
                                                 const float* __restrict__ gt,
                                                 float* __restrict__ acc) {
  __shared__ float sP[S_ * PITCH_];
  __shared__ float sG[S_ * PITCH_];
  __shared__ float sAcc[8 * 256];

  const int b     = blockIdx.x >> 7;           // 128 chunks per batch
  const int chunk = blockIdx.x & (CHUNKS_ - 1);
  const size_t k0 = (size_t)chunk * KC_;
  const int tid   = threadIdx.x;

  // Coalesced b128 staging of both 8x512 tiles into LDS.
  for (int r = 0; r < 4; ++r) {
    int idx = r * 256 + tid;       // float4 index 0..1023
    int s   = idx >> 7;            // 128 float4 per slot row
    int jj  = (idx & 127) << 2;    // float offset in row
    size_t g = ((size_t)(b * S_ + s) << 16) + k0 + (size_t)jj;
    float4 vp = *(const float4*)(pred + g);
    float4 vg = *(const float4*)(gt + g);
    *(float4*)&sP[s * PITCH_ + jj] = vp;
    *(float4*)&sG[s * PITCH_ + jj] = vg;
  }
  __syncthreads();

  const int wave  = tid >> 5;
  const int lane  = tid & 31;
  const int t     = lane & 15;     // M for A-frag, N for B-frag
  const int hi    = lane >> 4;     // K-group (adds 2 to K)
  const int kbase = wave * (KC_ / 8);

  v8f c = {};
  for (int i = 0; i < (KC_ / 8) / 4; ++i) {
    const int kk = kbase + 4 * i + 2 * hi;
    const int ra = (t & 7) * PITCH_ + kk;      // clamped in-bounds read
    float2 av = *(const float2*)&sP[ra];
    float2 bv = *(const float2*)&sG[ra];
    const float one = (t == 8) ? 1.0f : 0.0f;  // ones row/col at index 8
    v2f a, bb;
    a.x  = (t < 8) ? av.x : one;
    a.y  = (t < 8) ? av.y : one;
    bb.x = (t < 8) ? bv.x : one;
    bb.y = (t < 8) ? bv.y : one;
    c = __builtin_amdgcn_wmma_f32_16x16x4_f32(
        /*neg_a=*/false, a, /*neg_b=*/false, bb,
        /*c_mod=*/(short)0, c, /*reuse_a=*/false, /*reuse_b=*/false);
  }

  // Per-wave 16x16 partials -> LDS -> per-batch global accumulator.
#pragma unroll
  for (int v = 0; v < 8; ++v) {
    int M = v + 8 * hi;                        // C layout: VGPR v = row v / v+8
    sAcc[wave * 256 + M * 16 + t] = c[v];
  }
  __syncthreads();
  float sum = 0.0f;
  for (int w = 0; w < 8; ++w) sum += sAcc[w * 256 + tid];
  __hip_atomic_fetch_add(&acc[b * 256 + tid], sum,
                         __ATOMIC_RELAXED, __HIP_MEMORY_SCOPE_AGENT);
}

// ---------------------------------------------------------------- finalize --
__global__ void msl_finalize(const float* __restrict__ acc,
                             const int* __restrict__ nobj,
                             float* __restrict__ out) {
  __shared__ float sbg[32], sfg[32];
  __shared__ int scnt[32];
  const int b = threadIdx.x;  // one thread per batch, blockDim = 32

  float Mm[9][9];
  for (int p = 0; p < 9; ++p)
    for (int g = 0; g < 9; ++g)
      Mm[p][g] = acc[b * 256 + p * 16 + g];

  float bg = 1.0f - (2.0f * Mm[0][0] + EPS_) / (Mm[0][8] + Mm[8][0] + EPS_);

  float dice[7][7];
  for (int p = 0; p < 7; ++p)
    for (int g = 0; g < 7; ++g)
      dice[p][g] = (2.0f * Mm[p + 1][g + 1] + EPS_) /
                   (Mm[p + 1][8] + Mm[8][g + 1] + EPS_);

  int n = nobj[b];
  if (n > 7) n = 7;
  float fgs = 0.0f;
  int cnt = 0;
  if (n > 0) {
    const int mm = 7;                 // cols = pred slots; rows = n gt slots
    float cost[7][7];
    for (int i = 0; i < n; ++i)
      for (int j = 0; j < mm; ++j)
        cost[i][j] = -dice[j][i];     // transposed (n <= 7 rows)
    const float INF = 1e18f;
    float u[8], v[8];
    int pj[8], way[8];
    for (int j = 0; j <= mm; ++j) { u[j] = 0; v[j] = 0; pj[j] = 0; way[j] = 0; }
    for (int i = 1; i <= n; ++i) {
      pj[0] = i;
      int j0 = 0;
      float minv[8];
      bool used[8];
      for (int j = 0; j <= mm; ++j) { minv[j] = INF; used[j] = false; }
      while (true) {
        used[j0] = true;
        int i0 = pj[j0];
        float delta = INF;
        int j1 = 0;
        for (int j = 1; j <= mm; ++j)
          if (!used[j]) {
            float cur = cost[i0 - 1][j - 1] - u[i0] - v[j];
            if (cur < minv[j]) { minv[j] = cur; way[j] = j0; }
            if (minv[j] < delta) { delta = minv[j]; j1 = j; }
          }
        for (int j = 0; j <= mm; ++j) {
          if (used[j]) { u[pj[j]] += delta; v[j] -= delta; }
          else minv[j] -= delta;
        }
        j0 = j1;
        if (pj[j0] == 0) break;
      }
      while (j0) { int j1 = way[j0]; pj[j0] = pj[j1]; j0 = j1; }
    }
    for (int j = 1; j <= mm; ++j)
      if (pj[j] != 0) {
        int gi = pj[j] - 1;  // gt slot
        int pi = j - 1;      // pred slot
        fgs += 1.0f - dice[pi][gi];
        cnt++;
      }
  }

  sbg[b] = bg; sfg[b] = fgs; scnt[b] = cnt;
  __syncthreads();
  if (b == 0) {
    float tb = 0.0f, tf = 0.0f;
    int tc = 0;
    for (int i = 0; i < 32; ++i) { tb += sbg[i]; tf += sfg[i]; tc += scnt[i]; }
    out[0] = tb / 32.0f + (tc > 0 ? tf / (float)tc : 0.0f);
  }
}

// ---------------------------------------------------------------- launch ----
extern "C" void kernel_launch(void* const* d_in, const int* in_sizes, int n_in,
                              void* d_out, int out_size, void* d_ws, size_t ws_size,
                              hipStream_t stream) {
  const float* pred = (const float*)d_in[0];
  const float* gt   = (const float*)d_in[1];
  const int*   nob  = (const int*)d_in[2];
  float* out = (float*)d_out;
  float* acc = (float*)d_ws;  // 32 * 256 floats = 32 KB

  msl_init<<<B_, 256, 0, stream>>>(acc);
  msl_accum<<<B_ * CHUNKS_, 256, 0, stream>>>(pred, gt, acc);
  msl_finalize<<<1, 32, 0, stream>>>(acc, nob, out);
}